// ProtoLayer_21758304321939
// MI455X (gfx1250) — compile-verified
//
#include <hip/hip_runtime.h>

// Problem constants (from setup_inputs: t=16, way=20, shot=5, query=15, c=1024)
#define T      16
#define WAY    20
#define WAYP   32          // way padded to 2 WMMA N-tiles
#define SHOT   5
#define WQ     300         // way*query
#define WS     100         // way*shot
#define C      1024
#define MTILES 19          // ceil(300/16)
#define EPS    1e-12f

typedef __attribute__((ext_vector_type(2))) float v2f;
typedef __attribute__((ext_vector_type(8))) float v8f;

// ---------------------------------------------------------------------------
// Kernel 1: prototypes.  grid = T*WAYP blocks, 256 threads.
// proto_pad[t][n][c] = normalize(mean over shots)  (zeros for n >= WAY)
// p_sq[t][n]         = ||proto_pad[t][n]||^2       (0 for n >= WAY)
// ---------------------------------------------------------------------------
__global__ void proto_kernel(const float* __restrict__ sup,
                             float* __restrict__ proto_pad,
                             float* __restrict__ p_sq)
{
    const int t   = blockIdx.x >> 5;   // / WAYP
    const int n   = blockIdx.x & 31;   // % WAYP
    const int tid = threadIdx.x;       // 0..255, 4 channels each

    __shared__ float red[8];
    __shared__ float s_inv;

    float4 m = make_float4(0.f, 0.f, 0.f, 0.f);
    float  ss = 0.f;
    if (n < WAY) {
        const float* base = sup + (((size_t)t * WS) + (size_t)n * SHOT) * C + tid * 4;
        float4 a = make_float4(0.f, 0.f, 0.f, 0.f);
        #pragma unroll
        for (int s = 0; s < SHOT; ++s) {
            float4 v = *(const float4*)(base + s * C);
            a.x += v.x; a.y += v.y; a.z += v.z; a.w += v.w;
        }
        const float inv_shot = 1.0f / (float)SHOT;
        m.x = a.x * inv_shot; m.y = a.y * inv_shot;
        m.z = a.z * inv_shot; m.w = a.w * inv_shot;
        ss = m.x * m.x + m.y * m.y + m.z * m.z + m.w * m.w;
    }

    // block reduction of sum-of-squares (wave32-aware)
    #pragma unroll
    for (int off = 16; off >= 1; off >>= 1) ss += __shfl_xor(ss, off);
    if ((tid & 31) == 0) red[tid >> 5] = ss;
    __syncthreads();
    if (tid == 0) {
        float tot = 0.f;
        #pragma unroll
        for (int i = 0; i < 8; ++i) tot += red[i];
        float denom = fmaxf(sqrtf(tot), EPS);
        float inv   = 1.0f / denom;
        s_inv = inv;
        p_sq[blockIdx.x] = (n < WAY) ? tot * inv * inv : 0.0f;
    }
    __syncthreads();

    const float inv = s_inv;
    float4 o = make_float4(m.x * inv, m.y * inv, m.z * inv, m.w * inv); // zeros if n>=WAY
    *(float4*)(proto_pad + (size_t)blockIdx.x * C + tid * 4) = o;
}

// ---------------------------------------------------------------------------
// Kernel 2: fused (query-norm + GEMM + distance epilogue).
// One wave32 per (t, 16-row M-tile).  D = A(16xK raw queries) x B(Kx32 protos)
// via V_WMMA_F32_16X16X4_F32, row ||x||^2 co-accumulated from the A fragments.
// ---------------------------------------------------------------------------
__global__ __launch_bounds__(32)
void proto_dist_kernel(const float* __restrict__ q,
                       const float* __restrict__ proto_pad,
                       const float* __restrict__ p_sq,
                       float* __restrict__ out)
{
    const int bid  = blockIdx.x;
    const int t    = bid / MTILES;
    const int mt   = bid % MTILES;
    const int lane = threadIdx.x;      // 0..31
    const int half = lane >> 4;        // K-pair select (ISA A/B layout)
    const int l15  = lane & 15;
    const int mbase = mt * 16;

    int lrow = mbase + l15;            // row this lane loads for A
    if (lrow > WQ - 1) lrow = WQ - 1;  // clamp OOB loads (stores predicated later)

    const float* arow = q + ((size_t)t * WQ + lrow) * C + 2 * half;
    const int col0 = l15;              // N-tile 0 column
    const int col1 = 16 + l15;         // N-tile 1 column (valid only < WAY)
    const float* b0p = proto_pad + ((size_t)t * WAYP + col0) * C + 2 * half;
    const float* b1p = proto_pad + ((size_t)t * WAYP + col1) * C + 2 * half;

    v8f acc0 = {};
    v8f acc1 = {};
    float nacc = 0.f;                  // per-lane partial ||x||^2

    #pragma unroll 8
    for (int k = 0; k < C; k += 4) {
        v2f a  = *(const v2f*)(arow + k);
        v2f b0 = *(const v2f*)(b0p + k);
        v2f b1 = *(const v2f*)(b1p + k);
        nacc += a.x * a.x + a.y * a.y;
        // 8 args: (neg_a, A, neg_b, B, c_mod, C, reuse_a, reuse_b)
        acc0 = __builtin_amdgcn_wmma_f32_16x16x4_f32(
                   false, a, false, b0, (short)0, acc0, false, false);
        acc1 = __builtin_amdgcn_wmma_f32_16x16x4_f32(
                   false, a, false, b1, (short)0, acc1, false, false);
    }

    // combine the two K-halves: lanes j and j+16 both hold row (mbase+j) norm
    float nsum = nacc + __shfl_xor(nacc, 16);

    const float psq0 = p_sq[t * WAYP + col0];
    const float psq1 = p_sq[t * WAYP + col1];

    // D layout: VGPR r -> row (r + 8*half), col l15 (tile0) / 16+l15 (tile1)
    #pragma unroll
    for (int r = 0; r < 8; ++r) {
        const int rrow = mbase + r + 8 * half;
        float ss    = __shfl(nsum, r + 8 * half);       // ||x||^2 of rrow
        float denom = fmaxf(sqrtf(ss), EPS);
        float inv   = 1.0f / denom;
        float qsq   = ss * inv * inv;
        if (rrow < WQ) {
            float* o = out + ((size_t)t * WQ + rrow) * WAY;
            o[col0] = 2.0f * acc0[r] * inv - qsq - psq0;
            if (col1 < WAY)
                o[col1] = 2.0f * acc1[r] * inv - qsq - psq1;
        }
    }
}

// ---------------------------------------------------------------------------
extern "C" void kernel_launch(void* const* d_in, const int* in_sizes, int n_in,
                              void* d_out, int out_size, void* d_ws, size_t ws_size,
                              hipStream_t stream)
{
    const float* query_feat   = (const float*)d_in[0]; // [T, WQ, C]
    const float* support_feat = (const float*)d_in[1]; // [T, WS, C]
    float* out = (float*)d_out;                        // [T, WQ, WAY]

    // workspace: padded normalized prototypes + per-proto squared norms
    float* proto_pad = (float*)d_ws;                   // T*WAYP*C floats (2 MB)
    float* p_sq      = proto_pad + (size_t)T * WAYP * C; // T*WAYP floats

    proto_kernel<<<T * WAYP, 256, 0, stream>>>(support_feat, proto_pad, p_sq);
    proto_dist_kernel<<<T * MTILES, 32, 0, stream>>>(query_feat, proto_pad, p_sq, out);
}